// EdgeDecoder_37357625541095
// MI455X (gfx1250) — compile-verified
//
#include <hip/hip_runtime.h>
#include <hip/hip_bf16.h>

typedef float v2f __attribute__((ext_vector_type(2)));
typedef float v8f __attribute__((ext_vector_type(8)));

#define HCH 128
#define WAVES_PER_WG 4
#define WG_THREADS (WAVES_PER_WG * 32)
#define NUM_WGS 1280

// ---------------------------------------------------------------------------
// Precompute A = W_ui @ W1[:128,:] and B = W_iu @ W1[128:,:] into an
// INTERLEAVED pair-fragment layout (one ds_load_b128 per lane yields both
// WMMA B-operands):
//   pair-frag p = kk*8 + nt   (kk = K-step 0..31, nt = N-tile 0..7)
//   lane l owns floats [p*128 + l*4 .. +3] = (A_k, A_k+1, B_k, B_k+1)
//   where k = kk*4 + (l<16 ? 0 : 2) and the N column is nt*16 + (l&15).
// Element (i = K index, n = N index):
//   kk=i>>2, r=i&3, nt=n>>4, l=((r>>1)<<4)+(n&15), slot=r&1
//   A -> (kk*8+nt)*128 + l*4 + slot ;  B -> same + 2
// ---------------------------------------------------------------------------
__global__ void precompute_mats_kernel(const float* __restrict__ W_ui,
                                       const float* __restrict__ W_iu,
                                       const float* __restrict__ W1,
                                       float* __restrict__ wsAB) {
    const int i = blockIdx.x;    // K index of fused matrix (0..127)
    const int n = threadIdx.x;   // N index (0..127)
    float a = 0.0f, b = 0.0f;
    for (int j = 0; j < HCH; ++j) {
        a += W_ui[i * HCH + j] * W1[j * HCH + n];
        b += W_iu[i * HCH + j] * W1[(HCH + j) * HCH + n];
    }
    const int kk   = i >> 2;
    const int r    = i & 3;
    const int nt   = n >> 4;
    const int l    = ((r >> 1) << 4) + (n & 15);
    const int slot = r & 1;
    const int off  = (kk * 8 + nt) * 128 + l * 4 + slot;
    wsAB[off]     = a;
    wsAB[off + 2] = b;
}

// Fused bias: c[n] = b1[n] + sum_j b_ui[j]*W1[j][n] + b_iu[j]*W1[128+j][n]
__global__ void precompute_bias_kernel(const float* __restrict__ b_ui,
                                       const float* __restrict__ b_iu,
                                       const float* __restrict__ b1,
                                       const float* __restrict__ W1,
                                       float* __restrict__ wsC) {
    const int n = threadIdx.x;
    float c = b1[n];
    for (int j = 0; j < HCH; ++j) {
        c += b_ui[j] * W1[j * HCH + n];
        c += b_iu[j] * W1[(HCH + j) * HCH + n];
    }
    wsC[n] = c;
}

// ---------------------------------------------------------------------------
// Main kernel: one wave per 16-edge tile (grid-stride). Interleaved weight
// fragments in LDS; each K-step batches 8 ds_load_b128 (distinct register
// quads, all-immediate offsets) then issues 16 v_wmma_f32_16x16x4_f32 in two
// passes (user, movie) so dependent WMMAs on the same accumulator are 8
// instructions apart. Epilogue: bias+ReLU+dot(W2) + half-wave shfl reduction.
// ---------------------------------------------------------------------------
__global__ void __launch_bounds__(WG_THREADS)
edge_decoder_main_kernel(const float* __restrict__ z_user,
                         const float* __restrict__ z_movie,
                         const int* __restrict__ eidx,
                         const float* __restrict__ wsAB,  // 32768 floats (pair frags)
                         const float* __restrict__ wsC,   // 128 floats fused bias
                         const float* __restrict__ W2,    // 128 floats
                         const float* __restrict__ b2,    // 1 float
                         float* __restrict__ out,
                         int n_edges, int n_tiles, int total_waves) {
    extern __shared__ float lds[];   // 32768 floats = 128 KB pair-fragments

    // Cooperative LDS fill (contiguous, b128 transfers).
    {
        const float4* src = (const float4*)wsAB;
        float4* dst = (float4*)lds;
        for (int i = threadIdx.x; i < (2 * HCH * HCH) / 4; i += WG_THREADS)
            dst[i] = src[i];
    }
    __syncthreads();

    const int lane  = threadIdx.x & 31;
    const int wave  = threadIdx.x >> 5;
    const int l15   = lane & 15;
    const int khalf = lane >> 4;             // 0: K+{0,1}, 1: K+{2,3}
    const int gwave = blockIdx.x * WAVES_PER_WG + wave;

    // Per-lane epilogue constants (depend only on lane&15): fused bias + W2.
    float creg[8], w2reg[8];
#pragma unroll
    for (int nt = 0; nt < 8; ++nt) {
        creg[nt]  = wsC[nt * 16 + l15];
        w2reg[nt] = W2[nt * 16 + l15];
    }
    const float b2v = b2[0];
    const float* fb0 = lds + lane * 4;       // lane-local fragment base

    for (int tile = gwave; tile < n_tiles; tile += total_waves) {
        const int e0 = tile * 16;
        int e = e0 + l15;
        if (e >= n_edges) e = n_edges - 1;           // safe gather for tail
        const int er = eidx[e];                      // user row index
        const int ec = eidx[n_edges + e];            // movie row index
        const float* pu = z_user  + (size_t)er * HCH + khalf * 2;
        const float* pm = z_movie + (size_t)ec * HCH + khalf * 2;

        v8f acc[8];
#pragma unroll
        for (int nt = 0; nt < 8; ++nt)
#pragma unroll
            for (int j = 0; j < 8; ++j) acc[nt][j] = 0.0f;

#pragma unroll 2
        for (int kk = 0; kk < 32; ++kk) {
            // A-operands: 16 gathered rows, 4 K-values per step (ISA A layout).
            const v2f au = *(const v2f*)(pu + kk * 4);
            const v2f am = *(const v2f*)(pm + kk * 4);
            const float* fb = fb0 + kk * 1024;       // one pair-frag row per K-step

            // Batch all 8 fragment loads -> 8 ds_load_b128 in flight.
            float4 w[8];
#pragma unroll
            for (int nt = 0; nt < 8; ++nt)
                w[nt] = *(const float4*)(fb + nt * 128);

            // Pass 1: user GEMM (A-operand au, B-operand = w[nt].xy).
#pragma unroll
            for (int nt = 0; nt < 8; ++nt) {
                v2f bu; bu[0] = w[nt].x; bu[1] = w[nt].y;
                acc[nt] = __builtin_amdgcn_wmma_f32_16x16x4_f32(
                    false, au, false, bu, (short)0, acc[nt], false, false);
            }
            // Pass 2: movie GEMM (A-operand am, B-operand = w[nt].zw).
#pragma unroll
            for (int nt = 0; nt < 8; ++nt) {
                v2f bm; bm[0] = w[nt].z; bm[1] = w[nt].w;
                acc[nt] = __builtin_amdgcn_wmma_f32_16x16x4_f32(
                    false, am, false, bm, (short)0, acc[nt], false, false);
            }
        }

        // Epilogue: hidden = relu(acc + c); partial = hidden * W2[n]; reduce n.
        // acc[nt][r] @ lane l : edge M = r + (l>=16 ? 8 : 0), channel n = nt*16 + (l&15)
        float s[8];
#pragma unroll
        for (int r = 0; r < 8; ++r) s[r] = 0.0f;
#pragma unroll
        for (int nt = 0; nt < 8; ++nt) {
            const float cv = creg[nt], wv = w2reg[nt];
#pragma unroll
            for (int r = 0; r < 8; ++r) {
                float h = acc[nt][r] + cv;
                h = (h > 0.0f) ? h : 0.0f;
                s[r] += h * wv;
            }
        }
        // Reduce across the 16 lanes of each half-wave (wave32, width 16).
#pragma unroll
        for (int r = 0; r < 8; ++r) {
            float v = s[r];
            v += __shfl_xor(v, 8, 16);
            v += __shfl_xor(v, 4, 16);
            v += __shfl_xor(v, 2, 16);
            v += __shfl_xor(v, 1, 16);
            s[r] = v;
        }
        if (l15 == 0) {
            const int ebase = e0 + khalf * 8;       // lane0 -> edges 0..7, lane16 -> 8..15
#pragma unroll
            for (int r = 0; r < 8; ++r) {
                if (ebase + r < n_edges) out[ebase + r] = s[r] + b2v;
            }
        }
    }
}

// ---------------------------------------------------------------------------
// Launcher. Workspace layout (floats):
//   [0     .. 32767] interleaved A/B pair fragments
//   [32768 .. 32895] fused bias c
// ---------------------------------------------------------------------------
extern "C" void kernel_launch(void* const* d_in, const int* in_sizes, int n_in,
                              void* d_out, int out_size, void* d_ws, size_t ws_size,
                              hipStream_t stream) {
    const float* z_user  = (const float*)d_in[0];
    const float* z_movie = (const float*)d_in[1];
    const int*   eidx    = (const int*)  d_in[2];
    const float* W_ui    = (const float*)d_in[3];
    const float* b_ui    = (const float*)d_in[4];
    const float* W_iu    = (const float*)d_in[5];
    const float* b_iu    = (const float*)d_in[6];
    const float* W1      = (const float*)d_in[7];
    const float* b1      = (const float*)d_in[8];
    const float* W2      = (const float*)d_in[9];
    const float* b2      = (const float*)d_in[10];
    float* out = (float*)d_out;

    const int n_edges = in_sizes[2] / 2;
    const int n_tiles = (n_edges + 15) / 16;

    float* wsAB = (float*)d_ws;
    float* wsC  = wsAB + 2 * HCH * HCH;

    precompute_mats_kernel<<<HCH, HCH, 0, stream>>>(W_ui, W_iu, W1, wsAB);
    precompute_bias_kernel<<<1, HCH, 0, stream>>>(b_ui, b_iu, b1, W1, wsC);

    const int total_waves = NUM_WGS * WAVES_PER_WG;
    const size_t lds_bytes = 2 * HCH * HCH * sizeof(float);   // 128 KB
    edge_decoder_main_kernel<<<NUM_WGS, WG_THREADS, lds_bytes, stream>>>(
        z_user, z_movie, eidx, wsAB, wsC, W2, b2, out,
        n_edges, n_tiles, total_waves);
}